// PreintegrationModule_4681514353467
// MI455X (gfx1250) — compile-verified
//
#include <hip/hip_runtime.h>
#include <math.h>

// ---------------- configuration ----------------
#define T_STEPS   512
#define NCHANS    6
#define SEGS      16                     // segments per batch (parallel scan split)
#define SEG_LEN   (T_STEPS / SEGS)       // 32 steps per segment
#define TPB       128                    // threads per block (4 wave32)
#define BAT_PER_BLK (TPB / SEGS)         // 8 batches per block
#define TC        8                      // timesteps per chunk per segment
#define NCHUNK    (SEG_LEN / TC)         // 4 chunks
#define P_REG     12                     // 16B pieces per (batch,seg) region per chunk = TC*6*4/16
#define REG_DW    52                     // padded region stride in DWORDs (48 used, 16B aligned)
#define NREGION   TPB                    // one region per thread
#define TILE_DW   (NREGION * REG_DW)     // 6656 DWORDs per buffer (~26.6 KB)
#define P_THREAD  12                     // async pieces issued per thread per chunk

#define DT    0.005f                     // 1/200
#define DT2H  1.25e-5f                   // 0.5*dt*dt

// ---------------- CDNA5 async-to-LDS path (probe-guarded) ----------------
#if defined(__has_builtin)
#  if __has_builtin(__builtin_amdgcn_global_load_async_to_lds_b128)
#    define HAVE_ASYNC 1
#  endif
#endif
#ifndef HAVE_ASYNC
#  define HAVE_ASYNC 0
#endif

typedef int v4i __attribute__((ext_vector_type(4)));

#if HAVE_ASYNC
typedef __attribute__((address_space(1))) v4i gv4i;   // global 128-bit vector
typedef __attribute__((address_space(3))) v4i lv4i;   // LDS 128-bit vector
#  if __has_builtin(__builtin_amdgcn_s_wait_asynccnt)
#    define WAIT_ASYNC(n) __builtin_amdgcn_s_wait_asynccnt(n)
#  else
#    define WAIT_ASYNC(n) asm volatile("s_wait_asynccnt %0" ::"i"(n))
#  endif
#else
#  define WAIT_ASYNC(n)
#endif

__global__ __launch_bounds__(TPB) void preint_kernel(const float* __restrict__ in,
                                                     float* __restrict__ out, int B) {
  __shared__ float sbuf[2][TILE_DW];
  const int tid = threadIdx.x;
  const int b0  = blockIdx.x * BAT_PER_BLK;

  // Cooperative (async) fill of one double-buffered LDS tile:
  // tile = 8 batches x 16 segments x TC steps x 6 floats, region-padded.
  auto issue = [&](int c, int buf) {
#pragma unroll
    for (int i = 0; i < P_THREAD; ++i) {
      int q  = tid + i * TPB;         // piece id (lane-consecutive => coalesced)
      int r  = q / P_REG;             // (batch,seg) region 0..NREGION-1
      int o  = q - r * P_REG;         // 16B piece within region
      int bi = r >> 4;                // batch within block
      int s  = r & (SEGS - 1);        // segment
      int bg = b0 + bi; if (bg > B - 1) bg = B - 1;
      const float* g = in + (size_t)bg * (T_STEPS * NCHANS)
                          + (size_t)(s * SEG_LEN + c * TC) * NCHANS + o * 4;
      float* l = &sbuf[buf][r * REG_DW + o * 4];
#if HAVE_ASYNC
      __builtin_amdgcn_global_load_async_to_lds_b128((gv4i*)(void*)g, (lv4i*)(void*)l, 0, 0);
#else
      *(float4*)l = *(const float4*)g;
#endif
    }
  };

  // Segment-local state, started from identity: after SEG_LEN steps this IS the
  // segment's composition operator (A = dr, bv = dv, bp = dp).
  float r00 = 1.f, r01 = 0.f, r02 = 0.f;
  float r10 = 0.f, r11 = 1.f, r12 = 0.f;
  float r20 = 0.f, r21 = 0.f, r22 = 1.f;
  float dvx = 0.f, dvy = 0.f, dvz = 0.f;
  float dpx = 0.f, dpy = 0.f, dpz = 0.f;

  issue(0, 0);
  for (int c = 0; c < NCHUNK; ++c) {
    if (c + 1 < NCHUNK) issue(c + 1, (c + 1) & 1);
#if HAVE_ASYNC
    if (c + 1 < NCHUNK) { WAIT_ASYNC(P_THREAD); } else { WAIT_ASYNC(0); }
#endif
    __syncthreads();
    const float* base = &sbuf[c & 1][tid * REG_DW];
#pragma unroll
    for (int tt = 0; tt < TC; ++tt) {
      const float wxv = base[tt * 6 + 0];
      const float wyv = base[tt * 6 + 1];
      const float wzv = base[tt * 6 + 2];
      const float axv = base[tt * 6 + 3];
      const float ayv = base[tt * 6 + 4];
      const float azv = base[tt * 6 + 5];
      // Rodrigues: R = c I + s K + (1-c) u u^T
      float n2    = wxv * wxv + wyv * wyv + wzv * wzv;
      float rinv  = rsqrtf(n2);
      float theta = n2 * rinv * DT;
      float ux = wxv * rinv, uy = wyv * rinv, uz = wzv * rinv;
      float s   = __sinf(theta);
      float cth = __cosf(theta);
      float omc = 1.0f - cth;
      float xx = omc * ux * ux, yy = omc * uy * uy, zz = omc * uz * uz;
      float xy = omc * ux * uy, xz = omc * ux * uz, yz = omc * uy * uz;
      float sx = s * ux, sy = s * uy, sz = s * uz;
      float R00 = cth + xx, R01 = xy - sz, R02 = xz + sy;
      float R10 = xy + sz,  R11 = cth + yy, R12 = yz - sx;
      float R20 = xz - sy,  R21 = yz + sx,  R22 = cth + zz;
      // dr = dr @ R
      float t00 = r00 * R00 + r01 * R10 + r02 * R20;
      float t01 = r00 * R01 + r01 * R11 + r02 * R21;
      float t02 = r00 * R02 + r01 * R12 + r02 * R22;
      float t10 = r10 * R00 + r11 * R10 + r12 * R20;
      float t11 = r10 * R01 + r11 * R11 + r12 * R21;
      float t12 = r10 * R02 + r11 * R12 + r12 * R22;
      float t20 = r20 * R00 + r21 * R10 + r22 * R20;
      float t21 = r20 * R01 + r21 * R11 + r22 * R21;
      float t22 = r20 * R02 + r21 * R12 + r22 * R22;
      r00 = t00; r01 = t01; r02 = t02;
      r10 = t10; r11 = t11; r12 = t12;
      r20 = t20; r21 = t21; r22 = t22;
      // ra = dr_new @ a ; integrate (dp uses the UPDATED dv, per reference)
      float rax = r00 * axv + r01 * ayv + r02 * azv;
      float ray = r10 * axv + r11 * ayv + r12 * azv;
      float raz = r20 * axv + r21 * ayv + r22 * azv;
      dvx += rax * DT; dvy += ray * DT; dvz += raz * DT;
      dpx += dvx * DT + rax * DT2H;
      dpy += dvy * DT + ray * DT2H;
      dpz += dvz * DT + raz * DT2H;
    }
    __syncthreads();
  }

  // Combine the 16 segment operators (non-commutative tree reduce over lanes).
  // compose(L,R): A = A_L A_R ; bv = bv_L + A_L bv_R ; bp = bp_L + tauR*bv_L + A_L bp_R
#pragma unroll
  for (int rd = 0; rd < 4; ++rd) {
    const int   off  = 1 << rd;
    const float tauR = (float)(SEG_LEN << rd) * DT;
    float A00 = __shfl_down(r00, off, SEGS), A01 = __shfl_down(r01, off, SEGS), A02 = __shfl_down(r02, off, SEGS);
    float A10 = __shfl_down(r10, off, SEGS), A11 = __shfl_down(r11, off, SEGS), A12 = __shfl_down(r12, off, SEGS);
    float A20 = __shfl_down(r20, off, SEGS), A21 = __shfl_down(r21, off, SEGS), A22 = __shfl_down(r22, off, SEGS);
    float bvx = __shfl_down(dvx, off, SEGS), bvy = __shfl_down(dvy, off, SEGS), bvz = __shfl_down(dvz, off, SEGS);
    float bpx = __shfl_down(dpx, off, SEGS), bpy = __shfl_down(dpy, off, SEGS), bpz = __shfl_down(dpz, off, SEGS);
    float npx = dpx + tauR * dvx + r00 * bpx + r01 * bpy + r02 * bpz;
    float npy = dpy + tauR * dvy + r10 * bpx + r11 * bpy + r12 * bpz;
    float npz = dpz + tauR * dvz + r20 * bpx + r21 * bpy + r22 * bpz;
    float nvx = dvx + r00 * bvx + r01 * bvy + r02 * bvz;
    float nvy = dvy + r10 * bvx + r11 * bvy + r12 * bvz;
    float nvz = dvz + r20 * bvx + r21 * bvy + r22 * bvz;
    float n00 = r00 * A00 + r01 * A10 + r02 * A20;
    float n01 = r00 * A01 + r01 * A11 + r02 * A21;
    float n02 = r00 * A02 + r01 * A12 + r02 * A22;
    float n10 = r10 * A00 + r11 * A10 + r12 * A20;
    float n11 = r10 * A01 + r11 * A11 + r12 * A21;
    float n12 = r10 * A02 + r11 * A12 + r12 * A22;
    float n20 = r20 * A00 + r21 * A10 + r22 * A20;
    float n21 = r20 * A01 + r21 * A11 + r22 * A21;
    float n22 = r20 * A02 + r21 * A12 + r22 * A22;
    dpx = npx; dpy = npy; dpz = npz;
    dvx = nvx; dvy = nvy; dvz = nvz;
    r00 = n00; r01 = n01; r02 = n02;
    r10 = n10; r11 = n11; r12 = n12;
    r20 = n20; r21 = n21; r22 = n22;
  }

  // Lane with segment 0 holds the full-sequence (dr, dv, dp).
  if ((tid & (SEGS - 1)) == 0) {
    int bg = b0 + (tid >> 4);
    if (bg < B) {
      float tr = r00 + r11 + r22;
      float ca = 0.5f * (tr - 1.0f);
      ca = fminf(fmaxf(ca, -1.0f + 1e-6f), 1.0f - 1e-6f);
      float ang   = acosf(ca);
      float inv2s = 0.5f / sinf(ang);
      float axx = (r21 - r12) * inv2s;
      float axy = (r02 - r20) * inv2s;
      float axz = (r10 - r01) * inv2s;
      float h  = 0.5f * ang;
      float sh = sinf(h), chh = cosf(h);
      float* o = out + (size_t)bg * 7;
      o[0] = dpx; o[1] = dpy; o[2] = dpz;
      o[3] = chh; o[4] = axx * sh; o[5] = axy * sh; o[6] = axz * sh;
    }
  }
}

extern "C" void kernel_launch(void* const* d_in, const int* in_sizes, int n_in,
                              void* d_out, int out_size, void* d_ws, size_t ws_size,
                              hipStream_t stream) {
  (void)n_in; (void)out_size; (void)d_ws; (void)ws_size;
  const float* in = (const float*)d_in[0];
  float* out = (float*)d_out;
  const int B = in_sizes[0] / (T_STEPS * NCHANS);
  const int blocks = (B + BAT_PER_BLK - 1) / BAT_PER_BLK;
  preint_kernel<<<blocks, TPB, 0, stream>>>(in, out, B);
}